// SRenderY_37391985279423
// MI455X (gfx1250) — compile-verified
//
#include <hip/hip_runtime.h>
#include <hip/hip_bf16.h>
#include <cstddef>

#define BB 8
#define VV 5023
#define FF 9976
#define HH 512
#define WW 512
#define UVD 256
#define HWSZ (HH*WW)

typedef float v2f __attribute__((ext_vector_type(2)));
typedef float v8f __attribute__((ext_vector_type(8)));

// ---- output offsets (flat concat in reference return order) ----
#define O_IMG   ((size_t)0)         // images            (B,3,H,W)
#define O_ALB   ((size_t)6291456)   // albedo_images     (B,3,H,W)
#define O_ALPHA ((size_t)12582912)  // alpha_images      (B,1,H,W)
#define O_POS   ((size_t)14680064)  // pos_mask          (B,1,H,W)
#define O_SHAD  ((size_t)16777216)  // shading_images    (B,3,H,W)
#define O_GRID  ((size_t)23068672)  // grid              (B,H,W,2)
#define O_NRM   ((size_t)27262976)  // normals           (B,V,3)
#define O_NIMG  ((size_t)27383528)  // normal_images*a   (B,3,H,W)
#define O_TNRM  ((size_t)33674984)  // transformed_norms (B,V,3)

// SH constants (CF folded into basis)
#define CF0 0.28209479177387814f
#define CF1 1.0233267079464885f
#define CF4 0.8580855308201834f
#define CF7 0.4290427654100917f
#define CF8 0.24770795859447144f

__global__ void k_zero(float* __restrict__ p, int n) {
  int i = blockIdx.x * 256 + threadIdx.x;
  if (i < n) p[i] = 0.0f;
}

// one thread per (b, face): cross product scatter-add to the 3 vertices
__global__ void k_face_scatter(const float* __restrict__ verts,
                               const float* __restrict__ tverts,
                               const int*   __restrict__ faces,
                               float* accN, float* accT) {
  int i = blockIdx.x * 256 + threadIdx.x;
  if (i >= BB * FF) return;
  int b = i / FF, f = i - b * FF;
  int i0 = faces[f*3+0], i1 = faces[f*3+1], i2 = faces[f*3+2];

  const float* vb = verts  + (size_t)b * VV * 3;
  const float* tb = tverts + (size_t)b * VV * 3;

  // original vertices
  {
    float p0x=vb[i0*3+0], p0y=vb[i0*3+1], p0z=vb[i0*3+2];
    float p1x=vb[i1*3+0], p1y=vb[i1*3+1], p1z=vb[i1*3+2];
    float p2x=vb[i2*3+0], p2y=vb[i2*3+1], p2z=vb[i2*3+2];
    float e1x=p1x-p0x, e1y=p1y-p0y, e1z=p1z-p0z;
    float e2x=p2x-p0x, e2y=p2y-p0y, e2z=p2z-p0z;
    float cx = e1y*e2z - e1z*e2y;
    float cy = e1z*e2x - e1x*e2z;
    float cz = e1x*e2y - e1y*e2x;
    float* nb = accN + (size_t)b * VV * 3;
    atomicAdd(&nb[i0*3+0], cx); atomicAdd(&nb[i0*3+1], cy); atomicAdd(&nb[i0*3+2], cz);
    atomicAdd(&nb[i1*3+0], cx); atomicAdd(&nb[i1*3+1], cy); atomicAdd(&nb[i1*3+2], cz);
    atomicAdd(&nb[i2*3+0], cx); atomicAdd(&nb[i2*3+1], cy); atomicAdd(&nb[i2*3+2], cz);
  }
  // transformed vertices (uniform z-shift cancels in edge differences)
  {
    float p0x=tb[i0*3+0], p0y=tb[i0*3+1], p0z=tb[i0*3+2];
    float p1x=tb[i1*3+0], p1y=tb[i1*3+1], p1z=tb[i1*3+2];
    float p2x=tb[i2*3+0], p2y=tb[i2*3+1], p2z=tb[i2*3+2];
    float e1x=p1x-p0x, e1y=p1y-p0y, e1z=p1z-p0z;
    float e2x=p2x-p0x, e2y=p2y-p0y, e2z=p2z-p0z;
    float cx = e1y*e2z - e1z*e2y;
    float cy = e1z*e2x - e1x*e2z;
    float cz = e1x*e2y - e1y*e2x;
    float* nb = accT + (size_t)b * VV * 3;
    atomicAdd(&nb[i0*3+0], cx); atomicAdd(&nb[i0*3+1], cy); atomicAdd(&nb[i0*3+2], cz);
    atomicAdd(&nb[i1*3+0], cx); atomicAdd(&nb[i1*3+1], cy); atomicAdd(&nb[i1*3+2], cz);
    atomicAdd(&nb[i2*3+0], cx); atomicAdd(&nb[i2*3+1], cy); atomicAdd(&nb[i2*3+2], cz);
  }
}

// one thread per (b, vertex): normalize; write outputs 7 (normals) and 9 (transformed_normals)
__global__ void k_normalize(const float* __restrict__ accN,
                            const float* __restrict__ accT,
                            float* __restrict__ out) {
  int i = blockIdx.x * 256 + threadIdx.x;
  if (i >= BB * VV) return;
  {
    float x = accN[i*3+0], y = accN[i*3+1], z = accN[i*3+2];
    float inv = 1.0f / fmaxf(sqrtf(x*x + y*y + z*z), 1e-12f);
    out[O_NRM + (size_t)i*3+0] = x*inv;
    out[O_NRM + (size_t)i*3+1] = y*inv;
    out[O_NRM + (size_t)i*3+2] = z*inv;
  }
  {
    float x = accT[i*3+0], y = accT[i*3+1], z = accT[i*3+2];
    float inv = 1.0f / fmaxf(sqrtf(x*x + y*y + z*z), 1e-12f);
    out[O_TNRM + (size_t)i*3+0] = x*inv;
    out[O_TNRM + (size_t)i*3+1] = y*inv;
    out[O_TNRM + (size_t)i*3+2] = z*inv;
  }
}

// one thread per (b, face): build 36-float attribute record (16B aligned records)
__global__ void k_attr(const float* __restrict__ vertices,
                       const float* __restrict__ uvcoords,
                       const int*   __restrict__ faces,
                       const float* __restrict__ out,   // read normals back
                       float* __restrict__ attr) {
  int i = blockIdx.x * 256 + threadIdx.x;
  if (i >= BB * FF) return;
  int b = i / FF, f = i - b * FF;
  size_t base = (size_t)i * 36;
#pragma unroll
  for (int k = 0; k < 3; ++k) {
    int vi = faces[f*3 + k];
    size_t vb = ((size_t)b * VV + vi) * 3;
    size_t o = base + (size_t)k * 12;
    attr[o+0]  = uvcoords[f*9 + k*3 + 0];
    attr[o+1]  = uvcoords[f*9 + k*3 + 1];
    attr[o+2]  = uvcoords[f*9 + k*3 + 2];
    attr[o+3]  = out[O_TNRM + vb + 0];
    attr[o+4]  = out[O_TNRM + vb + 1];
    attr[o+5]  = out[O_TNRM + vb + 2];
    attr[o+6]  = vertices[vb + 0];
    attr[o+7]  = vertices[vb + 1];
    attr[o+8]  = vertices[vb + 2];
    attr[o+9]  = out[O_NRM + vb + 0];
    attr[o+10] = out[O_NRM + vb + 1];
    attr[o+11] = out[O_NRM + vb + 2];
  }
}

// fused per-pixel kernel: interp, grid, bilinear albedo, SH shading via WMMA f32 16x16x4
__global__ void __launch_bounds__(256)
k_main(const float* __restrict__ albedos,
       const float* __restrict__ lights,
       const float* __restrict__ bary,
       const int*   __restrict__ p2f,
       const float* __restrict__ attr,
       float* __restrict__ out) {
  __shared__ float sL[12 * 16];        // zero-padded lights: [k(12)][n(16)]
  __shared__ float sSH[8 * 32 * 12];   // per-wave SH basis:  [wave][pix32][k(12)]
  __shared__ float sSD[8 * 32 * 16];   // per-wave shading D: [wave][pix32][n(16)]

  const int tid  = threadIdx.x;
  const int lane = tid & 31;
  const int wave = tid >> 5;
  const int pid  = blockIdx.x * 256 + tid;       // grid divides evenly: no bounds branch
  const int b    = pid / HWSZ;                    // uniform within block (HW % 256 == 0)
  const int pix  = pid - b * HWSZ;

  // stage zero-padded lights (B matrix, K x N = 12 x 16) into LDS
  if (tid < 192) {
    int k = tid >> 4, n = tid & 15;
    float lv = 0.0f;
    if (k < 9 && n < 3) lv = lights[b*27 + k*3 + n];
    sL[tid] = lv;
  }

  // per-pixel gather + barycentric interpolation
  int face = p2f[pid];
  bool maskv = face < 0;
  int idx = maskv ? 0 : face;
  float b0 = bary[(size_t)pid*3+0];
  float b1 = bary[(size_t)pid*3+1];
  float b2 = bary[(size_t)pid*3+2];

  const float4* ap = (const float4*)(attr + (size_t)idx * 36);  // 144B records, 16B aligned
  float a[36];
#pragma unroll
  for (int i = 0; i < 9; ++i) {
    float4 q = ap[i];
    a[i*4+0] = q.x; a[i*4+1] = q.y; a[i*4+2] = q.z; a[i*4+3] = q.w;
  }
  float v[12];
#pragma unroll
  for (int d = 0; d < 12; ++d) {
    float t = b0*a[d] + b1*a[12+d] + b2*a[24+d];
    v[d] = maskv ? 0.0f : t;
  }
  float alpha = maskv ? 0.0f : 1.0f;

  // SH basis (CF folded), padded K to 12, into LDS (A matrix rows = pixels)
  {
    float Nx = v[9], Ny = v[10], Nz = v[11];
    int wl = (wave*32 + lane) * 12;
    sSH[wl+0]  = CF0;
    sSH[wl+1]  = CF1 * Nx;
    sSH[wl+2]  = CF1 * Ny;
    sSH[wl+3]  = CF1 * Nz;
    sSH[wl+4]  = CF4 * Nx * Ny;
    sSH[wl+5]  = CF4 * Nx * Nz;
    sSH[wl+6]  = CF4 * Ny * Nz;
    sSH[wl+7]  = CF7 * (Nx*Nx - Ny*Ny);
    sSH[wl+8]  = CF8 * (3.0f*Nz*Nz - 1.0f);
    sSH[wl+9]  = 0.0f;
    sSH[wl+10] = 0.0f;
    sSH[wl+11] = 0.0f;
  }
  __syncthreads();

  // D(16 pix x 16) = A(16x4 SH) x B(4x16 lights), K = 12 in 3 chained steps.
  // f32 operand layout: K split across lane halves (lanes 0-15: K0/K1, lanes 16-31: K2/K3);
  // B/D row-striped across lanes, rows r and r+8 in the two lane halves.
  {
    const int n     = lane & 15;
    const int khalf = (lane < 16) ? 0 : 2;
    const int rhalf = (lane < 16) ? 0 : 8;
#pragma unroll
    for (int p = 0; p < 2; ++p) {
      v8f acc = {0.0f,0.0f,0.0f,0.0f,0.0f,0.0f,0.0f,0.0f};
      const int pa = (wave*32 + p*16 + (lane & 15)) * 12;
#pragma unroll
      for (int kc = 0; kc < 12; kc += 4) {
        v2f av; av.x = sSH[pa + kc + khalf];
                av.y = sSH[pa + kc + khalf + 1];
        v2f bv; bv.x = sL[(kc + khalf) * 16 + n];
                bv.y = sL[(kc + khalf + 1) * 16 + n];
        acc = __builtin_amdgcn_wmma_f32_16x16x4_f32(
            /*neg_a=*/false, av, /*neg_b=*/false, bv,
            /*c_mod=*/(short)0, acc, /*reuse_a=*/false, /*reuse_b=*/false);
      }
#pragma unroll
      for (int r = 0; r < 8; ++r) {
        int pr = p*16 + r + rhalf;
        sSD[(wave*32 + pr) * 16 + n] = acc[r];
      }
    }
  }
  __syncthreads();

  float shade0 = sSD[(wave*32 + lane)*16 + 0];
  float shade1 = sSD[(wave*32 + lane)*16 + 1];
  float shade2 = sSD[(wave*32 + lane)*16 + 2];

  // grid, alpha, pos_mask
  float gx = v[0], gy = v[1];
  out[O_GRID + (size_t)pid*2 + 0] = gx;
  out[O_GRID + (size_t)pid*2 + 1] = gy;
  out[O_ALPHA + (size_t)pid] = alpha;
  out[O_POS   + (size_t)pid] = (v[5] > 0.05f) ? 1.0f : 0.0f;

  // bilinear albedo sample with border-zero validity (matches reference grid_sample)
  float xs = (gx + 1.0f) * (UVD * 0.5f) - 0.5f;
  float ys = (gy + 1.0f) * (UVD * 0.5f) - 0.5f;
  float x0f = floorf(xs), y0f = floorf(ys);
  int x0 = (int)x0f, y0 = (int)y0f;
  int x1 = x0 + 1,  y1 = y0 + 1;
  float fx = xs - x0f, fy = ys - y0f;
  float w00 = (1.0f-fx)*(1.0f-fy), w10 = fx*(1.0f-fy);
  float w01 = (1.0f-fx)*fy,        w11 = fx*fy;
  float mx0 = (x0 >= 0 && x0 < UVD) ? 1.0f : 0.0f;
  float mx1 = (x1 >= 0 && x1 < UVD) ? 1.0f : 0.0f;
  float my0 = (y0 >= 0 && y0 < UVD) ? 1.0f : 0.0f;
  float my1 = (y1 >= 0 && y1 < UVD) ? 1.0f : 0.0f;
  int cx0 = x0 < 0 ? 0 : (x0 > UVD-1 ? UVD-1 : x0);
  int cx1 = x1 < 0 ? 0 : (x1 > UVD-1 ? UVD-1 : x1);
  int cy0 = y0 < 0 ? 0 : (y0 > UVD-1 ? UVD-1 : y0);
  int cy1 = y1 < 0 ? 0 : (y1 > UVD-1 ? UVD-1 : y1);
  const float* img = albedos + (size_t)b * 3 * UVD * UVD;

  float shade[3] = {shade0, shade1, shade2};
#pragma unroll
  for (int c = 0; c < 3; ++c) {
    const float* ic = img + (size_t)c * UVD * UVD;
    float s00 = ic[cy0*UVD + cx0] * (mx0*my0);
    float s10 = ic[cy0*UVD + cx1] * (mx1*my0);
    float s01 = ic[cy1*UVD + cx0] * (mx0*my1);
    float s11 = ic[cy1*UVD + cx1] * (mx1*my1);
    float alb = s00*w00 + s10*w10 + s01*w01 + s11*w11;
    size_t co = ((size_t)b*3 + c) * HWSZ + pix;
    out[O_ALB  + co] = alb * alpha;
    out[O_IMG  + co] = alb * shade[c] * alpha;
    out[O_SHAD + co] = shade[c];
    out[O_NIMG + co] = v[9 + c] * alpha;
  }
}

extern "C" void kernel_launch(void* const* d_in, const int* in_sizes, int n_in,
                              void* d_out, int out_size, void* d_ws, size_t ws_size,
                              hipStream_t stream) {
  const float* vertices  = (const float*)d_in[0];
  const float* tvertices = (const float*)d_in[1];
  const float* albedos   = (const float*)d_in[2];
  const float* lights    = (const float*)d_in[3];
  const float* uvcoords  = (const float*)d_in[4];
  const float* bary      = (const float*)d_in[5];
  const int*   faces     = (const int*)d_in[6];
  const int*   p2f       = (const int*)d_in[7];
  float* out = (float*)d_out;

  float* w    = (float*)d_ws;
  float* accN = w;                               // B*V*3
  float* accT = w + (size_t)BB*VV*3;             // B*V*3
  float* attr = w + (size_t)2*BB*VV*3;           // B*F*36 (records 16B aligned)

  int nAcc = 2 * BB * VV * 3;
  k_zero<<<(nAcc + 255)/256, 256, 0, stream>>>(w, nAcc);

  int nBF = BB * FF;
  k_face_scatter<<<(nBF + 255)/256, 256, 0, stream>>>(vertices, tvertices, faces, accN, accT);

  int nBV = BB * VV;
  k_normalize<<<(nBV + 255)/256, 256, 0, stream>>>(accN, accT, out);

  k_attr<<<(nBF + 255)/256, 256, 0, stream>>>(vertices, uvcoords, faces, out, attr);

  k_main<<<(BB * HWSZ) / 256, 256, 0, stream>>>(albedos, lights, bary, p2f, attr, out);
}